// CRF_9216999817242
// MI455X (gfx1250) — compile-verified
//
#include <hip/hip_runtime.h>

#define B_TOT 256
#define T_LEN 1024
#define LBLS  128
#define NUMC  126
#define BT    16      // batch tile per workgroup
#define RPAD  4       // p_t row pad (floats): 132-float stride -> 64 distinct banks

typedef float v2f __attribute__((ext_vector_type(2)));
typedef float v8f __attribute__((ext_vector_type(8)));

// ---------------------------------------------------------------------------
// Kernel 1: gold path score per sequence. One wave32 per batch.
// ---------------------------------------------------------------------------
__global__ __launch_bounds__(256) void crf_gold_kernel(
    const float* __restrict__ logits, const int* __restrict__ labels,
    const int* __restrict__ lens, const float* __restrict__ trans,
    float* __restrict__ out)
{
    const int wave = threadIdx.x >> 5;
    const int lane = threadIdx.x & 31;
    const int b    = blockIdx.x * 8 + wave;

    const int len = lens[b];
    const float* lg = logits + (size_t)b * T_LEN * NUMC;
    const int*   lb = labels + (size_t)b * T_LEN;

    float acc = 0.0f;
    for (int t = lane; t < len; t += 32)          // unary
        acc += lg[t * NUMC + lb[t]];
    for (int e = lane; e <= len; e += 32) {       // binary (trans[cur, prev])
        int prev = (e == 0)   ? (LBLS - 2) : lb[e - 1];
        int cur  = (e == len) ? (LBLS - 1) : lb[e];
        acc += trans[cur * LBLS + prev];
    }
    #pragma unroll
    for (int off = 16; off > 0; off >>= 1)
        acc += __shfl_down(acc, off, 32);
    if (lane == 0) out[b] = acc;
}

// ---------------------------------------------------------------------------
// Kernel 2: forward normalizer, scaled-exp domain + f32 WMMA.
//   alpha[k,b] = C[b] + log p[k,b];  E = exp(trans) hoisted into A-fragments.
//   Per step: S = E @ P (32x v_wmma_f32_16x16x4_f32), q = exp(logit)*S,
//   renormalize by per-batch max, all in the WMMA D-register layout
//   (lane owns states 16*wave+8*hi+v of batch n). 2 barriers/step, f64 C.
// ---------------------------------------------------------------------------
__global__ __launch_bounds__(256) void crf_fwd_kernel(
    const float* __restrict__ logits, const int* __restrict__ lens,
    const float* __restrict__ trans, float* __restrict__ out)
{
    // p, batch-major with pad: B-frag b64 loads hit 64 distinct banks.
    __shared__ __attribute__((aligned(16))) float p_t[BT][LBLS + RPAD];
    __shared__ float red[16][BT];        // per-batch partial max / sums
    __shared__ int   len_lds[BT];

    const int tid   = threadIdx.x;
    const int wave  = tid >> 5;          // 0..7: state tile 16*wave..+15
    const int lane  = tid & 31;
    const int n     = lane & 15;         // batch column (WMMA N / B,D lanes)
    const int hi    = lane >> 4;         // half-wave select
    const int jbase = wave * 16 + hi * 8;// first state row owned by this lane
    const int rrow  = (wave << 1) | hi;  // this lane-group's row in red[]
    const int b0    = blockIdx.x * BT;

    // ---- A fragments: E[j,:] = exp(trans[j,:]) for this wave's 16 rows.
    // A layout (16x4 f32): lanes0-15 M=0..15; v0:K={0|2}, v1:K={1|3} by half.
    const int arow = wave * 16 + n;
    v2f afrag[32];
    #pragma unroll
    for (int kk = 0; kk < 32; ++kk) {
        const int col = kk * 4 + hi * 2;
        afrag[kk].x = __expf(trans[arow * LBLS + col]);
        afrag[kk].y = __expf(trans[arow * LBLS + col + 1]);
    }
    // ---- exp(trans[end, j]) for the final reduction (registers).
    float eend[8];
    #pragma unroll
    for (int v = 0; v < 8; ++v)
        eend[v] = __expf(trans[(LBLS - 1) * LBLS + jbase + v]);

    // ---- Init: p = onehot(start), cache lens.
    if (tid < BT) len_lds[tid] = lens[b0 + tid];
    #pragma unroll
    for (int v = 0; v < 8; ++v)
        p_t[n][jbase + v] = (jbase + v == LBLS - 2) ? 1.0f : 0.0f;
    __syncthreads();

    const int    mylen = len_lds[n];
    double       Creg  = 0.0;            // valid on tid<16 (wave0, hi0, n=tid)
    const float* lgp   = logits + (size_t)(b0 + n) * T_LEN * NUMC;

    // Branch-free clamped offsets for the two pad states (j >= NUMC).
    int off[8];
    #pragma unroll
    for (int v = 0; v < 8; ++v)
        off[v] = jbase + ((jbase + v < NUMC) ? v : 0);

    for (int t = 0; t < T_LEN; ++t) {
        // ---- Unconditional logit loads, issued early (hide under WMMA).
        float lgv[8];
        #pragma unroll
        for (int v = 0; v < 8; ++v)
            lgv[v] = lgp[t * NUMC + off[v]];

        // ---- All 32 B fragments into VGPRs first (contiguous b64 loads),
        // then fence so the WMMA chain runs with a single dscnt wait.
        v2f bfrag[32];
        #pragma unroll
        for (int kk = 0; kk < 32; ++kk)
            bfrag[kk] = *(const v2f*)&p_t[n][4 * kk + 2 * hi];
        asm volatile("" ::: "memory");

        v8f acc0 = {};
        v8f acc1 = {};
        #pragma unroll
        for (int kk = 0; kk < 16; ++kk)
            acc0 = __builtin_amdgcn_wmma_f32_16x16x4_f32(
                false, afrag[kk], false, bfrag[kk], (short)0, acc0, false, false);
        #pragma unroll
        for (int kk = 16; kk < 32; ++kk)
            acc1 = __builtin_amdgcn_wmma_f32_16x16x4_f32(
                false, afrag[kk], false, bfrag[kk], (short)0, acc1, false, false);
        const v8f accv = acc0 + acc1;

        // ---- q = exp(logit) * S; pad states masked to 0 (cndmask, no flow).
        float q[8];
        float lmax = 0.0f;
        #pragma unroll
        for (int v = 0; v < 8; ++v) {
            float qq = __expf(lgv[v]) * accv[v];
            qq = (jbase + v < NUMC) ? qq : 0.0f;
            q[v] = qq;
            lmax = fmaxf(lmax, qq);
        }
        red[rrow][n] = lmax;
        __syncthreads();

        // ---- Every thread reduces its batch's max (broadcast LDS reads).
        float m = red[0][n];
        #pragma unroll
        for (int r = 1; r < 16; ++r) m = fmaxf(m, red[r][n]);

        if (t < mylen) {
            const float inv = __builtin_amdgcn_rcpf(m);  // scale is arbitrary
            *(float4*)&p_t[n][jbase] =
                make_float4(q[0] * inv, q[1] * inv, q[2] * inv, q[3] * inv);
            *(float4*)&p_t[n][jbase + 4] =
                make_float4(q[4] * inv, q[5] * inv, q[6] * inv, q[7] * inv);
        }
        if (tid < BT && t < mylen)   // one lane per batch advances C (f64)
            Creg += (double)__logf(m);
        __syncthreads();
    }

    // ---- Final: norm = C + log( sum_j p[j] * exp(trans[end, j]) )
    float part = 0.0f;
    #pragma unroll
    for (int v = 0; v < 8; ++v)
        part += p_t[n][jbase + v] * eend[v];
    red[rrow][n] = part;
    __syncthreads();
    if (tid < BT) {
        float s = 0.0f;
        #pragma unroll
        for (int r = 0; r < 16; ++r) s += red[r][n];
        const float norm = (float)(Creg + (double)__logf(s));
        out[b0 + n] -= norm;   // gold - norm
    }
}

// ---------------------------------------------------------------------------
extern "C" void kernel_launch(void* const* d_in, const int* in_sizes, int n_in,
                              void* d_out, int out_size, void* d_ws, size_t ws_size,
                              hipStream_t stream) {
    const float* logits = (const float*)d_in[0];
    const int*   labels = (const int*)d_in[1];
    const int*   lens   = (const int*)d_in[2];
    const float* trans  = (const float*)d_in[3];
    float* out = (float*)d_out;

    crf_gold_kernel<<<B_TOT / 8, 256, 0, stream>>>(logits, labels, lens, trans, out);
    crf_fwd_kernel<<<B_TOT / BT, 256, 0, stream>>>(logits, lens, trans, out);
}